// DescartesExtension_5428838662289
// MI455X (gfx1250) — compile-verified
//
#include <hip/hip_runtime.h>

typedef __attribute__((ext_vector_type(2))) float v2f;
typedef __attribute__((ext_vector_type(8))) float v8f;

namespace {
constexpr int kD     = 1024;
constexpr int kNComb = kD * (kD + 1) / 2;  // 524800 outputs per batch row
constexpr int kTiles = kD / 16;            // 64 tiles per dimension
}

// One wave computes one 16x16 tile of the per-row outer product x x^T via
// V_WMMA_F32_16X16X4_F32 (K=0 holds the data, K=1..3 zero-padded; fp32 exact),
// then stores into the flat upper-triangular layout:
//   flat(i,j) = i*D - i*(i-1)/2 + (j - i),  i <= j
// Full tiles (tj > ti) use incremental addressing: addr(i+1) = addr(i) + D-1-i.
// Diagonal tiles exploit symmetry: lanes with j < i store the identical value
// to flat(j,i) -- unconditional stores, no EXEC-mask branches.
__global__ __launch_bounds__(128)
void descartes_outer_wmma(const float* __restrict__ x,
                          float* __restrict__ out,
                          int batch) {
    const int wave = threadIdx.x >> 5;       // 4 waves per block
    const int lane = threadIdx.x & 31;
    const int tj   = blockIdx.x * 4 + wave;  // tile column (j/16)
    const int ti   = blockIdx.y;             // tile row    (i/16)
    if (tj < ti) return;                     // uniform per wave
    const int b = blockIdx.z;
    if (b >= batch) return;

    const int half = lane >> 4;              // 0: lanes 0-15, 1: lanes 16-31
    const int n    = lane & 15;
    const int i0   = ti << 4;
    const int j0   = tj << 4;

    const float* __restrict__ xrow = x + b * kD;
    // Unconditional loads (4KB row, L1/L2 resident), lane-select keeps EXEC
    // all-ones into the WMMA.
    const float xi = xrow[i0 + n];
    const float xj = xrow[j0 + n];

    // A: 16x4 fp32; lanes 0-15 hold {K=0,K=1}, lanes 16-31 {K=2,K=3}. K=0 live.
    v2f a;
    a.x = half ? 0.0f : xi;
    a.y = 0.0f;
    // B: 4x16 fp32, row-striped across lanes; K=0 live.
    v2f bmat;
    bmat.x = half ? 0.0f : xj;
    bmat.y = 0.0f;

    v8f c = {};
    // (neg_a, A, neg_b, B, c_mod, C, reuse_a, reuse_b)
    v8f d = __builtin_amdgcn_wmma_f32_16x16x4_f32(
        false, a, false, bmat, (short)0, c, false, false);

    // D layout: VGPR r holds M = r (lanes 0-15) and M = r+8 (lanes 16-31).
    const unsigned rowBase = (unsigned)b * (unsigned)kNComb;
    const int j      = j0 + n;
    const int iFirst = i0 + (half << 3);

    if (tj > ti) {
        // Hot path (~97% of tiles): j > i always. Incremental flat addressing.
        unsigned addr = rowBase
                      + (unsigned)iFirst * (unsigned)kD
                      - (unsigned)((iFirst * (iFirst - 1)) >> 1)
                      + (unsigned)(j - iFirst);
        unsigned step = (unsigned)(kD - 1 - iFirst);
#pragma unroll
        for (int r = 0; r < 8; ++r) {
            __builtin_nontemporal_store(d[r], out + addr);  // NT: 537MB stream >> 192MB L2
            addr += step;
            step -= 1u;
        }
    } else {
        // Diagonal tile: symmetric redirect, all stores unconditional.
#pragma unroll
        for (int r = 0; r < 8; ++r) {
            const int i  = iFirst + r;
            const int lo = (i < j) ? i : j;
            const int hi = (i < j) ? j : i;
            const unsigned off = (unsigned)lo * (unsigned)kD
                               - (unsigned)((lo * (lo - 1)) >> 1)
                               + (unsigned)(hi - lo);
            __builtin_nontemporal_store(d[r], out + rowBase + off);
        }
    }
}

extern "C" void kernel_launch(void* const* d_in, const int* in_sizes, int n_in,
                              void* d_out, int out_size, void* d_ws, size_t ws_size,
                              hipStream_t stream) {
    (void)n_in; (void)out_size; (void)d_ws; (void)ws_size;
    const float* x = (const float*)d_in[0];
    float* out     = (float*)d_out;
    const int batch = in_sizes[0] / kD;  // 256

    dim3 block(128);                       // 4 waves, one 16x16 tile each
    dim3 grid(kTiles / 4, kTiles, batch);  // (16, 64, 256)
    hipLaunchKernelGGL(descartes_outer_wmma, grid, block, 0, stream, x, out, batch);
}